// QNN_25125558681859
// MI455X (gfx1250) — compile-verified
//
#include <hip/hip_runtime.h>
#include <hip/hip_bf16.h>
#include <math.h>

typedef float v2f __attribute__((ext_vector_type(2)));
typedef float v8f __attribute__((ext_vector_type(8)));

#define DIM   4096      // 2^12 amplitudes
#define BLOCK 128       // 4 waves of 32 (wave32)

__device__ __forceinline__ float2 cmul(float2 a, float2 b) {
    return make_float2(a.x*b.x - a.y*b.y, a.x*b.y + a.y*b.x);
}

// Apply a 16x16 complex matrix G (row-major float2[256] in LDS) to one nibble of
// the state. Element address: idx = q*sq + t*st + n*sn (q = contracted row,
// t = tile 0..15, n = column-in-tile). Each wave owns 4 tiles; read/write
// regions per wave are disjoint, so no barrier is needed inside.
__device__ void apply16(float2* S, const float2* G, int sq, int st, int sn,
                        int lane, int wave) {
    const int half = lane >> 4;       // 0 or 1
    const int lm   = lane & 15;

    // A fragments: 16x4 f32 layout — lanes0-15: VGPR0=K0, VGPR1=K1;
    // lanes16-31: VGPR0=K2, VGPR1=K3. Four K-blocks cover K=16.
    v2f Ar[4], Ai[4], nAi[4];
#pragma unroll
    for (int kb = 0; kb < 4; ++kb) {
#pragma unroll
        for (int j = 0; j < 2; ++j) {
            int k = kb*4 + j + 2*half;
            float2 gv = G[lm*16 + k];
            Ar[kb][j]  = gv.x;
            Ai[kb][j]  = gv.y;
            nAi[kb][j] = -gv.y;
        }
    }

#pragma unroll
    for (int i = 0; i < 4; ++i) {
        int t  = wave*4 + i;
        int bt = t * st;
        // B fragments: 4x16 block, VGPR j holds rows j (lanes0-15) and j+2 (lanes16-31)
        v2f Br[4], Bi[4];
#pragma unroll
        for (int kb = 0; kb < 4; ++kb) {
#pragma unroll
            for (int j = 0; j < 2; ++j) {
                int q = kb*4 + j + 2*half;
                float2 sv = S[q*sq + bt + lm*sn];
                Br[kb][j] = sv.x;
                Bi[kb][j] = sv.y;
            }
        }
        v8f accR = {0,0,0,0,0,0,0,0};
        v8f accI = {0,0,0,0,0,0,0,0};
        // Re(D) = Gr*Sr - Gi*Si ; Im(D) = Gr*Si + Gi*Sr  (K=16 via 4 chained x4 WMMAs)
#pragma unroll
        for (int kb = 0; kb < 4; ++kb)
            accR = __builtin_amdgcn_wmma_f32_16x16x4_f32(false, Ar[kb],  false, Br[kb], (short)0, accR, false, false);
#pragma unroll
        for (int kb = 0; kb < 4; ++kb)
            accR = __builtin_amdgcn_wmma_f32_16x16x4_f32(false, nAi[kb], false, Bi[kb], (short)0, accR, false, false);
#pragma unroll
        for (int kb = 0; kb < 4; ++kb)
            accI = __builtin_amdgcn_wmma_f32_16x16x4_f32(false, Ar[kb],  false, Bi[kb], (short)0, accI, false, false);
#pragma unroll
        for (int kb = 0; kb < 4; ++kb)
            accI = __builtin_amdgcn_wmma_f32_16x16x4_f32(false, Ai[kb],  false, Br[kb], (short)0, accI, false, false);
        // D/C layout: VGPR v -> rows v (lanes0-15) and v+8 (lanes16-31), N = lane%16
#pragma unroll
        for (int v = 0; v < 8; ++v) {
            int m = v + 8*half;
            S[m*sq + bt + lm*sn] = make_float2(accR[v], accI[v]);
        }
    }
}

// tensorcircuit r-gate 2x2 (theta, phi, alpha)
__device__ __forceinline__ void stage_r(float2 M[2][2], float th, float ph, float al) {
    float ct = cosf(th), st = sinf(th);
    float sa = sinf(al);
    float nx = sa * cosf(ph), ny = sa * sinf(ph), nz = cosf(al);
    M[0][0] = make_float2(ct,     -st*nz);
    M[0][1] = make_float2(-st*ny, -st*nx);
    M[1][0] = make_float2( st*ny, -st*nx);
    M[1][1] = make_float2(ct,      st*nz);
}

// rx gate 2x2: exp(-i theta/2 X)
__device__ __forceinline__ void stage_rx(float2 M[2][2], float th) {
    float c = cosf(0.5f*th), s = sinf(0.5f*th);
    M[0][0] = make_float2(c, 0.f);
    M[0][1] = make_float2(0.f, -s);
    M[1][0] = make_float2(0.f, -s);
    M[1][1] = make_float2(c, 0.f);
}

// G[p,q] = prod_o M2[o][bit_o(p)][bit_o(q)]  (o=0 is MSB of the nibble)
__device__ __forceinline__ void fill_kron(float2* G, const float2 (*M2)[2][2], int tid) {
    for (int e = tid; e < 256; e += BLOCK) {
        int p = e >> 4, q = e & 15;
        float2 prod = M2[0][(p>>3)&1][(q>>3)&1];
        prod = cmul(prod, M2[1][(p>>2)&1][(q>>2)&1]);
        prod = cmul(prod, M2[2][(p>>1)&1][(q>>1)&1]);
        prod = cmul(prod, M2[3][ p     &1][ q     &1]);
        G[e] = prod;
    }
}

// Fold lifted RXX (in-group, qubit offsets o,o+1) into G: rows p and p^m couple:
// a' = c*a - i*s*b ; b' = c*b - i*s*a.  128 pairs, one per thread.
__device__ __forceinline__ void fold_rxx(float2* G, float th, int o, int tid) {
    float c = cosf(0.5f*th), s = sinf(0.5f*th);
    int hb  = 3 - o;
    int msk = (1 << (3-o)) | (1 << (2-o));
    int q   = tid & 15;
    int jj  = tid >> 4;                                   // 0..7
    int p   = ((jj >> hb) << (hb+1)) | (jj & ((1<<hb)-1)); // canonical: bit hb = 0
    int p2  = p ^ msk;
    float2 a = G[p*16+q], b = G[p2*16+q];
    G[p*16+q]  = make_float2(c*a.x + s*b.y, c*a.y - s*b.x);
    G[p2*16+q] = make_float2(c*b.x + s*a.y, c*b.y - s*a.x);
}

// Cross-group RXX applied directly to the state: pairs {p, p^msk}.
__device__ __forceinline__ void cross_rxx(float2* S, float th, int msk, int hb, int tid) {
    float c = cosf(0.5f*th), s = sinf(0.5f*th);
#pragma unroll
    for (int it = 0; it < (DIM/2)/BLOCK; ++it) {          // 16 pairs/thread
        int jj = tid + BLOCK*it;                          // 0..2047
        int p  = ((jj >> hb) << (hb+1)) | (jj & ((1<<hb)-1));
        int p2 = p ^ msk;
        float2 a = S[p], b = S[p2];
        S[p]  = make_float2(c*a.x + s*b.y, c*a.y - s*b.x);
        S[p2] = make_float2(c*b.x + s*a.y, c*b.y - s*a.x);
    }
}

__global__ __launch_bounds__(BLOCK)
void qnn_kernel(const float* __restrict__ x,     // (512,12)
                const float* __restrict__ fmin_p,
                const float* __restrict__ w,     // (2,2,12,3)
                const float* __restrict__ ew,    // (2,2,11)
                const float* __restrict__ fr,    // (12,3)
                const float* __restrict__ dk,    // (1,1)
                const float* __restrict__ db,    // (1,)
                float* __restrict__ out)         // (512,1)
{
    __shared__ float2 S[DIM];            // 32 KB state
    __shared__ float2 G[256];            // fused 16x16 gate
    __shared__ float2 M2[4][2][2];       // staged per-qubit 2x2s
    __shared__ float  RED[BLOCK];

    const int tid  = threadIdx.x;
    const int b    = blockIdx.x;
    const int lane = tid & 31;
    const int wave = tid >> 5;
    const float fmin = fmin_p[0];

    // strides for the three qubit-nibbles (axis 0 = qubits 0-3 = top nibble)
    const int SQ[3] = {256, 16, 1};
    const int ST[3] = {16, 256, 256};
    const int SN[3] = {1, 1, 16};

    // |0...0>
    for (int i = tid; i < DIM; i += BLOCK)
        S[i] = make_float2(i == 0 ? 1.f : 0.f, 0.f);
    __syncthreads();

    for (int f = 0; f < 2; ++f) {
        // data re-uploading RX layer (same xs each frequency)
        for (int g = 0; g < 3; ++g) {
            if (tid < 4)
                stage_rx(M2[tid], x[b*12 + 4*g + tid] * fmin);
            __syncthreads();
            fill_kron(G, M2, tid);
            __syncthreads();
            apply16(S, G, SQ[g], ST[g], SN[g], lane, wave);
            __syncthreads();
        }
        for (int k = 0; k < 2; ++k) {
            for (int g = 0; g < 3; ++g) {
                if (tid < 4) {
                    int j = 4*g + tid;
                    const float* wp = w + (((f*2 + k)*12 + j) * 3);
                    stage_r(M2[tid], wp[0], wp[1], wp[2]);   // theta=w0, phi=w1, alpha=w2
                }
                __syncthreads();
                fill_kron(G, M2, tid);
                __syncthreads();
                // in-group RXX (all RXX mutually commute)
                for (int o = 0; o < 3; ++o) {
                    fold_rxx(G, ew[(f*2 + k)*11 + 4*g + o], o, tid);
                    __syncthreads();
                }
                apply16(S, G, SQ[g], ST[g], SN[g], lane, wave);
                __syncthreads();
            }
            // cross-group RXX: j=3 (qubit bits 8,7) and j=7 (qubit bits 4,3)
            cross_rxx(S, ew[(f*2 + k)*11 + 3], 0x180, 8, tid);
            __syncthreads();
            cross_rxx(S, ew[(f*2 + k)*11 + 7], 0x018, 4, tid);
            __syncthreads();
        }
    }
    // final rotation layer
    for (int g = 0; g < 3; ++g) {
        if (tid < 4) {
            const float* fp = fr + (4*g + tid) * 3;
            stage_r(M2[tid], fp[0], fp[1], fp[2]);
        }
        __syncthreads();
        fill_kron(G, M2, tid);
        __syncthreads();
        apply16(S, G, SQ[g], ST[g], SN[g], lane, wave);
        __syncthreads();
    }

    // <Z_0>: qubit 0 is bit 11 of the flat index
    float part = 0.f;
    for (int i = tid; i < DIM; i += BLOCK) {
        float2 a = S[i];
        float p2 = a.x*a.x + a.y*a.y;
        part += ((i >> 11) & 1) ? -p2 : p2;
    }
    RED[tid] = part;
    __syncthreads();
    if (tid == 0) {
        float z = 0.f;
        for (int i = 0; i < BLOCK; ++i) z += RED[i];
        out[b] = z * dk[0] + db[0];
    }
}

extern "C" void kernel_launch(void* const* d_in, const int* in_sizes, int n_in,
                              void* d_out, int out_size, void* d_ws, size_t ws_size,
                              hipStream_t stream) {
    const float* x    = (const float*)d_in[0];
    const float* fmin = (const float*)d_in[1];
    const float* w    = (const float*)d_in[2];
    const float* ew   = (const float*)d_in[3];
    const float* fr   = (const float*)d_in[4];
    const float* dk   = (const float*)d_in[5];
    const float* db   = (const float*)d_in[6];
    float* out = (float*)d_out;
    qnn_kernel<<<512, BLOCK, 0, stream>>>(x, fmin, w, ew, fr, dk, db, out);
}